// SpatialTemporalAttention_33535104647287
// MI455X (gfx1250) — compile-verified
//
#include <hip/hip_runtime.h>

#define Bn 32
#define Tn 32
#define Nn 800
#define Fin 16
#define Hn 64
#define NLAYERS 5
#define LP 32   // padded K-stride for p1/p2 (multiple of 4, >= max L = 31)

typedef float v2f __attribute__((ext_vector_type(2)));
typedef float v8f __attribute__((ext_vector_type(8)));

// eval-mode BN: x * g/sqrt(1+1e-5) + b
#define BN_INV 0.9999950000374997f

#define WMMA_F32(a, b, c) \
  __builtin_amdgcn_wmma_f32_16x16x4_f32(false, (a), false, (b), (short)0, (c), false, false)

// ---------------------------------------------------------------------------
// Feature mapping: x0[b,h,n,t] = bn_start(relu(X[b,t,n,:] @ fm_w[:,h] + fm_b[h]))
// grid (N, B), block 64 (one thread per h)
// ---------------------------------------------------------------------------
__global__ void fm_kernel(const float* __restrict__ X, const float* __restrict__ fw,
                          const float* __restrict__ fb, const float* __restrict__ sg,
                          const float* __restrict__ sb, float* __restrict__ x0) {
  int n = blockIdx.x, b = blockIdx.y, h = threadIdx.x;
  __shared__ float w[Fin][Hn];
  for (int f = 0; f < Fin; ++f) w[f][h] = fw[f * Hn + h];
  __syncthreads();
  float bias = fb[h];
  float g = sg[h] * BN_INV, bb = sb[h];
  for (int t = 0; t < Tn; ++t) {
    const float* xp = X + (((size_t)b * Tn + t) * Nn + n) * Fin;
    float acc = bias;
#pragma unroll
    for (int f = 0; f < Fin; ++f) acc += xp[f] * w[f][h];
    acc = fmaxf(acc, 0.0f) * g + bb;
    x0[(((size_t)b * Hn + h) * Nn + n) * Tn + t] = acc;
  }
}

// ---------------------------------------------------------------------------
// Fused dilated TCN conv (K=2) + ReLU + BN, and residual 1x1 conv (pre-sliced
// to the last Lout positions, i.e. offset by dil).
// grid (N, B), block 64 (one thread per output channel h)
// ---------------------------------------------------------------------------
__global__ void conv_kernel(const float* __restrict__ x_in,
                            const float* __restrict__ tw, const float* __restrict__ tb,
                            const float* __restrict__ tg, const float* __restrict__ tbe,
                            const float* __restrict__ rw, const float* __restrict__ rb,
                            float* __restrict__ x_out, float* __restrict__ res,
                            int Lin, int Lout, int dil) {
  int n = blockIdx.x, b = blockIdx.y, h = threadIdx.x;
  __shared__ float xs[Hn][Tn];
  __shared__ float w0[Hn][Hn], w1[Hn][Hn], wr[Hn][Hn];
  const float* xp = x_in + (((size_t)b * Hn + h) * Nn + n) * Lin;
  for (int l = 0; l < Lin; ++l) xs[h][l] = xp[l];
  for (int c = 0; c < Hn; ++c) {
    w0[h][c] = tw[(h * Hn + c) * 2 + 0];
    w1[h][c] = tw[(h * Hn + c) * 2 + 1];
    wr[h][c] = rw[h * Hn + c];
  }
  __syncthreads();
  float tbias = tb[h], g = tg[h] * BN_INV, be = tbe[h], rbias = rb[h];
  size_t obase = ((size_t)b * Hn + h) * Nn + n;
  for (int l = 0; l < Lout; ++l) {
    float acc = tbias, racc = rbias;
    for (int c = 0; c < Hn; ++c) {
      acc += w0[h][c] * xs[c][l] + w1[h][c] * xs[c][l + dil];
      racc += wr[h][c] * xs[c][l + dil];
    }
    acc = fmaxf(acc, 0.0f) * g + be;
    x_out[obase * Lout + l] = acc;
    res[obase * Lout + l] = racc;
  }
}

// ---------------------------------------------------------------------------
// SAN projections into K-padded buffers (stride LP, zero-padded tail):
//   t1[f]      = bn1( sum_l x[b,f,n,l] * w1[l] )
//   p1[b,n,l]  = bn2( sum_f t1[f] * w2[l,f] )
//   p2[b,n,l]  = bn3( sum_f x[b,f,n,l] * w3[f] )
// grid (N, B), block 64
// ---------------------------------------------------------------------------
__global__ void p_kernel(const float* __restrict__ x,
                         const float* __restrict__ w1, const float* __restrict__ w2,
                         const float* __restrict__ w3,
                         const float* __restrict__ g1, const float* __restrict__ b1,
                         const float* __restrict__ g2, const float* __restrict__ b2,
                         const float* __restrict__ g3, const float* __restrict__ b3,
                         float* __restrict__ p1, float* __restrict__ p2, int L) {
  int n = blockIdx.x, b = blockIdx.y, f = threadIdx.x;
  __shared__ float t1[Hn];
  __shared__ float w1s[Tn];
  __shared__ float w3s[Hn];
  if (f < L) w1s[f] = w1[f];
  w3s[f] = w3[f];
  __syncthreads();
  const float* xp = x + (((size_t)b * Hn + f) * Nn + n) * L;
  float acc = 0.0f;
  for (int l = 0; l < L; ++l) acc += xp[l] * w1s[l];
  t1[f] = acc * (g1[n] * BN_INV) + b1[n];
  __syncthreads();
  size_t pbase = ((size_t)b * Nn + n) * LP;
  if (f < L) {
    int l = f;
    const float* w2r = w2 + l * Hn;
    float a2 = 0.0f;
    for (int c = 0; c < Hn; ++c) a2 += t1[c] * w2r[c];
    p1[pbase + l] = a2 * (g2[n] * BN_INV) + b2[n];
    float a3 = 0.0f;
    const float* xc = x + ((size_t)b * Hn * Nn + n) * L + l;  // stride N*L over c
    for (int c = 0; c < Hn; ++c) a3 += xc[(size_t)c * Nn * L] * w3s[c];
    p2[pbase + l] = a3 * (g3[n] * BN_INV) + b3[n];
  } else if (f < LP) {  // zero-fill K padding so the WMMA K-loop is unconditional
    p1[pbase + f] = 0.0f;
    p2[pbase + f] = 0.0f;
  }
}

// ---------------------------------------------------------------------------
// s[b,n,m] = relu( sum_l p1[b,n,l] * p2[b,m,l] ), K padded to LP=32.
// 2x2 register tiles per wave; unconditional aligned v2f loads.
// grid (25, 25, B), block 32
// ---------------------------------------------------------------------------
__global__ void s_wmma_kernel(const float* __restrict__ p1, const float* __restrict__ p2,
                              float* __restrict__ s) {
  int tn0 = blockIdx.x * 2, tm0 = blockIdx.y * 2, b = blockIdx.z;
  int lane = threadIdx.x & 31;
  int mi = lane & 15, kh = lane >> 4;
  const float* A0 = p1 + ((size_t)b * Nn + tm0 * 16 + mi) * LP;
  const float* A1 = A0 + 16 * LP;
  const float* B0 = p2 + ((size_t)b * Nn + tn0 * 16 + mi) * LP;
  const float* B1 = B0 + 16 * LP;
  v8f c00 = {}, c01 = {}, c10 = {}, c11 = {};
#pragma unroll
  for (int kk = 0; kk < LP; kk += 4) {
    int k0 = kk + 2 * kh;
    v2f a0 = *(const v2f*)(A0 + k0);
    v2f a1 = *(const v2f*)(A1 + k0);
    v2f b0 = *(const v2f*)(B0 + k0);
    v2f b1 = *(const v2f*)(B1 + k0);
    c00 = WMMA_F32(a0, b0, c00);
    c01 = WMMA_F32(a0, b1, c01);
    c10 = WMMA_F32(a1, b0, c10);
    c11 = WMMA_F32(a1, b1, c11);
  }
  float* out = s + (size_t)b * Nn * Nn + (size_t)(tm0 * 16) * Nn + tn0 * 16;
#pragma unroll
  for (int r = 0; r < 8; ++r) {
    int row = r + 8 * kh;
    out[row * Nn + mi]             = fmaxf(c00[r], 0.0f);
    out[row * Nn + 16 + mi]        = fmaxf(c01[r], 0.0f);
    out[(row + 16) * Nn + mi]      = fmaxf(c10[r], 0.0f);
    out[(row + 16) * Nn + 16 + mi] = fmaxf(c11[r], 0.0f);
  }
}

// ---------------------------------------------------------------------------
// attn_pre[b,n,k] = sum_m s[b,n,m] * v[k,m] + vb[k]   (K = 800)
// 2 row-tiles x 5 col-tiles per wave: 7 fragment loads -> 10 WMMAs per K-step
// (~11.4 FLOP/B from L2). grid (10, 25, B), block 32
// ---------------------------------------------------------------------------
__global__ void v_wmma_kernel(const float* __restrict__ s, const float* __restrict__ v,
                              const float* __restrict__ vb, float* __restrict__ attn) {
  int tn0 = blockIdx.x * 5, tm0 = blockIdx.y * 2, b = blockIdx.z;
  int lane = threadIdx.x & 31;
  int mi = lane & 15, kh = lane >> 4;
  const float* A0 = s + (size_t)b * Nn * Nn + (size_t)(tm0 * 16 + mi) * Nn;
  const float* A1 = A0 + 16 * Nn;
  const float* Bb = v + (size_t)(tn0 * 16 + mi) * Nn;
  v8f c0[5] = {}, c1[5] = {};
#pragma unroll 2
  for (int kk = 0; kk < Nn; kk += 4) {
    int k0 = kk + 2 * kh;
    v2f a0 = *(const v2f*)(A0 + k0);
    v2f a1 = *(const v2f*)(A1 + k0);
    v2f bf[5];
#pragma unroll
    for (int j = 0; j < 5; ++j) bf[j] = *(const v2f*)(Bb + (size_t)j * 16 * Nn + k0);
#pragma unroll
    for (int j = 0; j < 5; ++j) {
      c0[j] = WMMA_F32(a0, bf[j], c0[j]);
      c1[j] = WMMA_F32(a1, bf[j], c1[j]);
    }
  }
  float* out = attn + (size_t)b * Nn * Nn + (size_t)(tm0 * 16) * Nn + tn0 * 16;
#pragma unroll
  for (int j = 0; j < 5; ++j) {
    float bias = vb[(tn0 + j) * 16 + mi];
#pragma unroll
    for (int r = 0; r < 8; ++r) {
      int row = r + 8 * kh;
      out[row * Nn + j * 16 + mi]        = c0[j][r] + bias;
      out[(row + 16) * Nn + j * 16 + mi] = c1[j][r] + bias;
    }
  }
}

// ---------------------------------------------------------------------------
// Row softmax over last dim (800). grid (N, B), block 256.
// ---------------------------------------------------------------------------
__global__ void softmax_kernel(float* __restrict__ attn) {
  int n = blockIdx.x, b = blockIdx.y, t = threadIdx.x;
  float* row = attn + (size_t)b * Nn * Nn + (size_t)n * Nn;
  __shared__ float red[256];
  float mx = -1e30f;
  for (int i = t; i < Nn; i += 256) mx = fmaxf(mx, row[i]);
  red[t] = mx; __syncthreads();
  for (int st = 128; st > 0; st >>= 1) { if (t < st) red[t] = fmaxf(red[t], red[t + st]); __syncthreads(); }
  mx = red[0]; __syncthreads();
  float sum = 0.0f;
  for (int i = t; i < Nn; i += 256) { float e = __expf(row[i] - mx); row[i] = e; sum += e; }
  red[t] = sum; __syncthreads();
  for (int st = 128; st > 0; st >>= 1) { if (t < st) red[t] += red[t + st]; __syncthreads(); }
  float inv = 1.0f / red[0];
  for (int i = t; i < Nn; i += 256) row[i] *= inv;
}

// ---------------------------------------------------------------------------
// Aggregation + residual + BN (dominant GEMM, ~270 GFLOP total):
//   x_new[b,f,n,l] = bn( sum_m attn[b,n,m] * x[b,f,m,l] + res[b,f,n,l] )
// GEMM per b: M = n (800), cols = f*L + l (64L), K = m (800). 2x2 tiles/wave.
// grid (2L, 25, B), block 32
// ---------------------------------------------------------------------------
__global__ void agg_wmma_kernel(const float* __restrict__ attn, const float* __restrict__ x,
                                const float* __restrict__ res,
                                const float* __restrict__ bg, const float* __restrict__ bbe,
                                float* __restrict__ x_out, int L) {
  int tc0 = blockIdx.x * 2, tm0 = blockIdx.y * 2, b = blockIdx.z;
  int lane = threadIdx.x & 31;
  int mi = lane & 15, kh = lane >> 4;
  const float* A0 = attn + (size_t)b * Nn * Nn + (size_t)(tm0 * 16 + mi) * Nn;
  const float* A1 = A0 + 16 * Nn;
  int col0 = tc0 * 16 + mi, col1 = col0 + 16;
  int f0 = col0 / L, l0 = col0 % L;
  int f1 = col1 / L, l1 = col1 % L;
  const float* Bp0 = x + ((size_t)b * Hn + f0) * Nn * L + l0;  // + m*L
  const float* Bp1 = x + ((size_t)b * Hn + f1) * Nn * L + l1;
  v8f c00 = {}, c01 = {}, c10 = {}, c11 = {};
#pragma unroll 2
  for (int kk = 0; kk < Nn; kk += 4) {
    int k0 = kk + 2 * kh;
    v2f a0 = *(const v2f*)(A0 + k0);
    v2f a1 = *(const v2f*)(A1 + k0);
    v2f b0, b1;
    b0.x = Bp0[(size_t)k0 * L];
    b0.y = Bp0[(size_t)(k0 + 1) * L];
    b1.x = Bp1[(size_t)k0 * L];
    b1.y = Bp1[(size_t)(k0 + 1) * L];
    c00 = WMMA_F32(a0, b0, c00);
    c01 = WMMA_F32(a0, b1, c01);
    c10 = WMMA_F32(a1, b0, c10);
    c11 = WMMA_F32(a1, b1, c11);
  }
  float g0 = bg[f0] * BN_INV, be0 = bbe[f0];
  float g1 = bg[f1] * BN_INV, be1 = bbe[f1];
#pragma unroll
  for (int r = 0; r < 8; ++r) {
    int n0 = tm0 * 16 + r + 8 * kh;
    int n1 = n0 + 16;
    size_t i00 = (((size_t)b * Hn + f0) * Nn + n0) * L + l0;
    size_t i01 = (((size_t)b * Hn + f1) * Nn + n0) * L + l1;
    size_t i10 = (((size_t)b * Hn + f0) * Nn + n1) * L + l0;
    size_t i11 = (((size_t)b * Hn + f1) * Nn + n1) * L + l1;
    x_out[i00] = (c00[r] + res[i00]) * g0 + be0;
    x_out[i01] = (c01[r] + res[i01]) * g1 + be1;
    x_out[i10] = (c10[r] + res[i10]) * g0 + be0;
    x_out[i11] = (c11[r] + res[i11]) * g1 + be1;
  }
}

// ---------------------------------------------------------------------------
// out[b,n,h] = x[b,h,n,0]  (final L == 1)
// ---------------------------------------------------------------------------
__global__ void out_kernel(const float* __restrict__ x, float* __restrict__ out) {
  int i = blockIdx.x * 256 + threadIdx.x;
  if (i >= Bn * Nn * Hn) return;
  int h = i % Hn;
  int n = (i / Hn) % Nn;
  int b = i / (Nn * Hn);
  out[i] = x[((size_t)b * Hn + h) * Nn + n];
}

extern "C" void kernel_launch(void* const* d_in, const int* in_sizes, int n_in,
                              void* d_out, int out_size, void* d_ws, size_t ws_size,
                              hipStream_t stream) {
  // Input order (setup_inputs dict order, tuples flattened):
  const float* X       = (const float*)d_in[0];
  const float* fm_w    = (const float*)d_in[1];
  const float* fm_b    = (const float*)d_in[2];
  const float* start_g = (const float*)d_in[3];
  const float* start_b = (const float*)d_in[4];
  const float* tcn_w   = (const float*)d_in[5];
  const float* tcn_b   = (const float*)d_in[6];
  const float* tcn_g   = (const float*)d_in[7];
  const float* tcn_be  = (const float*)d_in[8];
  const float* res_w   = (const float*)d_in[9];
  const float* res_b   = (const float*)d_in[10];
  const float* bn_g    = (const float*)d_in[11];
  const float* bn_be   = (const float*)d_in[12];
  const float* san_w1[NLAYERS]; const float* san_w2[NLAYERS]; const float* san_w3[NLAYERS];
  const float* san_v[NLAYERS];  const float* san_vb[NLAYERS];
  for (int i = 0; i < NLAYERS; ++i) {
    san_w1[i] = (const float*)d_in[13 + i];
    san_w2[i] = (const float*)d_in[18 + i];
    san_w3[i] = (const float*)d_in[23 + i];
    san_v[i]  = (const float*)d_in[28 + i];
    san_vb[i] = (const float*)d_in[33 + i];
  }
  const float* san_g1 = (const float*)d_in[38];
  const float* san_b1 = (const float*)d_in[39];
  const float* san_g2 = (const float*)d_in[40];
  const float* san_b2 = (const float*)d_in[41];
  const float* san_g3 = (const float*)d_in[42];
  const float* san_b3 = (const float*)d_in[43];

  // Workspace carve-up (floats)
  float* ws = (float*)d_ws;
  size_t off = 0;
  float* xA   = ws + off; off += (size_t)Bn * Hn * Nn * Tn;   // ping
  float* xB   = ws + off; off += (size_t)Bn * Hn * Nn * Tn;   // pong
  float* resb = ws + off; off += (size_t)Bn * Hn * Nn * 31;
  float* sbuf = ws + off; off += (size_t)Bn * Nn * Nn;
  float* attn = ws + off; off += (size_t)Bn * Nn * Nn;
  float* p1   = ws + off; off += (size_t)Bn * Nn * LP;
  float* p2   = ws + off; off += (size_t)Bn * Nn * LP;
  (void)ws_size; (void)n_in; (void)in_sizes; (void)out_size;

  fm_kernel<<<dim3(Nn, Bn), 64, 0, stream>>>(X, fm_w, fm_b, start_g, start_b, xA);

  static const int dils[NLAYERS] = {1, 2, 4, 8, 16};
  int Lin = Tn;
  float* xin = xA;
  float* xtc = xB;
  for (int i = 0; i < NLAYERS; ++i) {
    int dil = dils[i];
    int Lout = Lin - dil;
    conv_kernel<<<dim3(Nn, Bn), 64, 0, stream>>>(
        xin, tcn_w + (size_t)i * Hn * Hn * 2, tcn_b + i * Hn, tcn_g + i * Hn, tcn_be + i * Hn,
        res_w + (size_t)i * Hn * Hn, res_b + i * Hn, xtc, resb, Lin, Lout, dil);
    p_kernel<<<dim3(Nn, Bn), 64, 0, stream>>>(
        xtc, san_w1[i], san_w2[i], san_w3[i],
        san_g1 + i * Nn, san_b1 + i * Nn, san_g2 + i * Nn, san_b2 + i * Nn,
        san_g3 + i * Nn, san_b3 + i * Nn, p1, p2, Lout);
    s_wmma_kernel<<<dim3(Nn / 32, Nn / 32, Bn), 32, 0, stream>>>(p1, p2, sbuf);
    v_wmma_kernel<<<dim3(Nn / 80, Nn / 32, Bn), 32, 0, stream>>>(sbuf, san_v[i], san_vb[i], attn);
    softmax_kernel<<<dim3(Nn, Bn), 256, 0, stream>>>(attn);
    agg_wmma_kernel<<<dim3(2 * Lout, Nn / 32, Bn), 32, 0, stream>>>(
        attn, xtc, resb, bn_g + i * Hn, bn_be + i * Hn, xin, Lout);
    // xin now holds the layer output (length Lout); xtc remains the scratch pong
    Lin = Lout;
  }
  out_kernel<<<(Bn * Nn * Hn + 255) / 256, 256, 0, stream>>>(xin, (float*)d_out);
}